// NonParamPoseEstimator_23407571763407
// MI455X (gfx1250) — compile-verified
//
#include <hip/hip_runtime.h>
#include <stdint.h>

// Problem constants (match reference)
#define Bb   64
#define Nn   131072
#define Pp   18
#define Gg   (Bb * Pp)          // 1152 groups
#define TILE 1024               // points staged per tile
#define TPB_STREAM 8            // tiles per block in streaming kernels
#define TILE3 (TILE * 3)        // floats of xyz per tile (3072)

// ---------------------------------------------------------------------------
// CDNA5 async global->LDS (lane-scatter) helpers.
// ---------------------------------------------------------------------------
__device__ __forceinline__ void async_ld_b128(void* lds_ptr, const void* gptr) {
    unsigned lds_off = (unsigned)(unsigned long long)lds_ptr;
    asm volatile("global_load_async_to_lds_b128 %0, %1, off"
                 :: "v"(lds_off), "v"(gptr)
                 : "memory");
}
__device__ __forceinline__ void wait_async0() {
    asm volatile("s_wait_asynccnt 0x0" ::: "memory");
}

// ---------------------------------------------------------------------------
// CDNA5 Tensor Data Mover: 1D contiguous f32 tile, global -> LDS.
// D# built per ISA ch.8: group0 = {count=1, lds_addr, global_addr, type=2},
// group1 = {data_size=4B, tensor_dim0=tile_dim0=nelem, dim1=1, stride=nelem},
// groups 2/3 zero (dims > 2 unused). Issue from one wave; EXEC is ignored by
// TDM. Completion via TENSORcnt.
// ---------------------------------------------------------------------------
typedef __attribute__((ext_vector_type(4))) unsigned v4u;
typedef __attribute__((ext_vector_type(8))) unsigned v8u;

__device__ __forceinline__ void tdm_load_1d_f32(void* lds_dst, const void* gsrc,
                                                unsigned nelem) {
    unsigned long long ga = (unsigned long long)gsrc;
    v4u g0;
    g0.x = 1u;                                        // count=1 (user descriptor)
    g0.y = (unsigned)(unsigned long long)lds_dst;     // lds_addr (bytes)
    g0.z = (unsigned)ga;                              // global_addr[31:0]
    g0.w = (unsigned)((ga >> 32) & 0x01FFFFFFu) | (2u << 30); // addr[56:32] | type=2
    v8u g1;
    g1.s0 = (2u << 16);                               // wg_mask=0, data_size=2 (4B)
    g1.s1 = (nelem & 0xFFFFu) << 16;                  // tensor_dim0[15:0]
    g1.s2 = (nelem >> 16) | (1u << 16);               // tensor_dim0[31:16], tensor_dim1=1
    g1.s3 = (nelem & 0xFFFFu) << 16;                  // tile_dim0 (fits 16 bits)
    g1.s4 = 1u;                                       // tile_dim1=1, tile_dim2=0
    g1.s5 = nelem;                                    // tensor_dim0_stride[31:0]
    g1.s6 = 0u;
    g1.s7 = 0u;
    v4u gz; gz.x = 0u; gz.y = 0u; gz.z = 0u; gz.w = 0u;
    asm volatile("tensor_load_to_lds %0, %1, %2, %3"
                 :: "s"(g0), "s"(g1), "s"(gz), "s"(gz)
                 : "memory");
}

// ---------------------------------------------------------------------------
// Kernel 1: per-group sums & counts. TDM-staged xyz tile + async label tile,
// LDS accumulators, one global-atomic flush per block.
// ---------------------------------------------------------------------------
__global__ __launch_bounds__(256)
void k_segsum(const float* __restrict__ xyz, const int* __restrict__ lab,
              float* __restrict__ gsum, unsigned* __restrict__ gcnt) {
    __shared__ float    lsum[Pp * 3];
    __shared__ unsigned lcnt[Pp];
    __shared__ float    sx[TILE3];
    __shared__ int      sl[TILE];
    const int tid   = threadIdx.x;
    const int bpb   = Nn / (TILE * TPB_STREAM);      // 16 blocks per batch
    const int b     = blockIdx.x / bpb;
    const int chunk = blockIdx.x % bpb;
    if (tid < Pp * 3) lsum[tid] = 0.0f;
    if (tid < Pp)     lcnt[tid] = 0u;
    __syncthreads();
    const size_t base = (size_t)b * Nn + (size_t)chunk * (TILE * TPB_STREAM);
    for (int t = 0; t < TPB_STREAM; ++t) {
        size_t tb = base + (size_t)t * TILE;
        if (tid < 32) {                               // wave 0 drives the TDM
            tdm_load_1d_f32(sx, xyz + tb * 3, TILE3);
            __builtin_amdgcn_s_wait_tensorcnt(0);
        }
        async_ld_b128(&sl[tid * 4], lab + tb + tid * 4);
        wait_async0();
        __syncthreads();
        for (int j = 0; j < 4; ++j) {
            int p = tid * 4 + j;
            int l = sl[p];
            atomicAdd(&lsum[l * 3 + 0], sx[p * 3 + 0]);
            atomicAdd(&lsum[l * 3 + 1], sx[p * 3 + 1]);
            atomicAdd(&lsum[l * 3 + 2], sx[p * 3 + 2]);
            atomicAdd(&lcnt[l], 1u);
        }
        __syncthreads();
    }
    if (tid < Pp * 3) atomicAdd(&gsum[b * Pp * 3 + tid], lsum[tid]);
    if (tid < Pp)     atomicAdd(&gcnt[b * Pp + tid], lcnt[tid]);
}

// ---------------------------------------------------------------------------
// Kernel 2: centers = sum / max(cnt,1); init radix-select state
// ---------------------------------------------------------------------------
__global__ void k_centers(const float* __restrict__ gsum, const unsigned* __restrict__ gcnt,
                          float* __restrict__ center, unsigned* __restrict__ selP,
                          unsigned* __restrict__ selK) {
    int g = blockIdx.x * blockDim.x + threadIdx.x;
    if (g >= Gg) return;
    unsigned c = gcnt[g];
    float f = (float)(c ? c : 1u);
    center[g * 3 + 0] = gsum[g * 3 + 0] / f;
    center[g * 3 + 1] = gsum[g * 3 + 1] / f;
    center[g * 3 + 2] = gsum[g * 3 + 2] / f;
    selP[g] = 0u;
    selK[g] = c ? ((c - 1u) >> 1) : 0u;    // 0-indexed rank of the median
}

// ---------------------------------------------------------------------------
// Kernel 3: distances (cached to ws, b128 stores) + level-0 histogram on
// bits[30:23]. TDM-staged xyz, async-staged labels.
// ---------------------------------------------------------------------------
__global__ __launch_bounds__(256)
void k_dist_hist0(const float* __restrict__ xyz, const int* __restrict__ lab,
                  const float* __restrict__ center, float* __restrict__ dist,
                  unsigned* __restrict__ hist) {
    __shared__ float    cc[Pp * 3];
    __shared__ unsigned lh[Pp * 256];
    __shared__ float    sx[TILE3];
    __shared__ int      sl[TILE];
    const int tid   = threadIdx.x;
    const int bpb   = Nn / (TILE * TPB_STREAM);
    const int b     = blockIdx.x / bpb;
    const int chunk = blockIdx.x % bpb;
    if (tid < Pp * 3) cc[tid] = center[b * Pp * 3 + tid];
    for (int i = tid; i < Pp * 256; i += 256) lh[i] = 0u;
    __syncthreads();
    const size_t base = (size_t)b * Nn + (size_t)chunk * (TILE * TPB_STREAM);
    for (int t = 0; t < TPB_STREAM; ++t) {
        size_t tb = base + (size_t)t * TILE;
        if (tid < 32) {
            tdm_load_1d_f32(sx, xyz + tb * 3, TILE3);
            __builtin_amdgcn_s_wait_tensorcnt(0);
        }
        async_ld_b128(&sl[tid * 4], lab + tb + tid * 4);
        wait_async0();
        __syncthreads();
        float4 dv;
        float* dvp = (float*)&dv;
        for (int j = 0; j < 4; ++j) {
            int p = tid * 4 + j;
            int l = sl[p];
            float dx = sx[p * 3 + 0] - cc[l * 3 + 0];
            float dy = sx[p * 3 + 1] - cc[l * 3 + 1];
            float dz = sx[p * 3 + 2] - cc[l * 3 + 2];
            float d = sqrtf(dx * dx + dy * dy + dz * dz);
            dvp[j] = d;
            unsigned bits = __float_as_uint(d);       // sign bit is 0
            atomicAdd(&lh[l * 256 + (bits >> 23)], 1u);
        }
        *(float4*)(dist + tb + tid * 4) = dv;         // coalesced b128 store
        __syncthreads();
    }
    for (int i = tid; i < Pp * 256; i += 256) {
        unsigned v = lh[i];
        if (v) atomicAdd(&hist[(size_t)b * Pp * 256 + i], v);
    }
}

// ---------------------------------------------------------------------------
// Kernel 4: radix-select scan, one thread per group.
// ---------------------------------------------------------------------------
__global__ void k_scan(const unsigned* __restrict__ hist, unsigned* __restrict__ selP,
                       unsigned* __restrict__ selK, float* __restrict__ median,
                       int shift, int nbins, int finalPass) {
    int g = blockIdx.x * blockDim.x + threadIdx.x;
    if (g >= Gg) return;
    unsigned k = selK[g], acc = 0, newk = 0;
    int chosen = nbins - 1;
    for (int bn = 0; bn < nbins; ++bn) {
        unsigned h = hist[(size_t)g * 256 + bn];
        if (acc + h > k) { chosen = bn; newk = k - acc; break; }
        acc += h;
    }
    unsigned p = selP[g] | ((unsigned)chosen << shift);
    selP[g] = p;
    selK[g] = newk;
    if (finalPass) median[g] = __uint_as_float(p);
}

// ---------------------------------------------------------------------------
// Kernel 5: histogram refinement over cached dist+labels, async-staged.
// ---------------------------------------------------------------------------
__global__ __launch_bounds__(256)
void k_hist_refine(const float* __restrict__ dist, const int* __restrict__ lab,
                   const unsigned* __restrict__ selP, unsigned* __restrict__ hist,
                   int shift, int prevshift, unsigned binmask) {
    __shared__ unsigned lh[Pp * 256];
    __shared__ unsigned pref[Pp];
    __shared__ float    sd[TILE];
    __shared__ int      sl[TILE];
    const int tid   = threadIdx.x;
    const int bpb   = Nn / (TILE * TPB_STREAM);
    const int b     = blockIdx.x / bpb;
    const int chunk = blockIdx.x % bpb;
    if (tid < Pp) pref[tid] = selP[b * Pp + tid] >> prevshift;
    for (int i = tid; i < Pp * 256; i += 256) lh[i] = 0u;
    __syncthreads();
    const size_t base = (size_t)b * Nn + (size_t)chunk * (TILE * TPB_STREAM);
    for (int t = 0; t < TPB_STREAM; ++t) {
        size_t tb = base + (size_t)t * TILE;
        async_ld_b128(&sd[tid * 4], dist + tb + tid * 4);
        async_ld_b128(&sl[tid * 4], lab + tb + tid * 4);
        wait_async0();
        __syncthreads();
        for (int j = 0; j < 4; ++j) {
            int p = tid * 4 + j;
            unsigned bits = __float_as_uint(sd[p]);
            int l = sl[p];
            if ((bits >> prevshift) == pref[l])
                atomicAdd(&lh[l * 256 + ((bits >> shift) & binmask)], 1u);
        }
        __syncthreads();
    }
    for (int i = tid; i < Pp * 256; i += 256) {
        unsigned v = lh[i];
        if (v) atomicAdd(&hist[(size_t)b * Pp * 256 + i], v);
    }
}

// ---------------------------------------------------------------------------
// Kernel 6: filtered sums (dist <= group median). TDM xyz + async dist/labels.
// ---------------------------------------------------------------------------
__global__ __launch_bounds__(256)
void k_final(const float* __restrict__ xyz, const float* __restrict__ dist,
             const int* __restrict__ lab, const float* __restrict__ median,
             float* __restrict__ fsum, unsigned* __restrict__ fcnt) {
    __shared__ float    med[Pp];
    __shared__ float    ls[Pp * 3];
    __shared__ unsigned lc[Pp];
    __shared__ float    sx[TILE3];
    __shared__ float    sd[TILE];
    __shared__ int      sl[TILE];
    const int tid   = threadIdx.x;
    const int bpb   = Nn / (TILE * TPB_STREAM);
    const int b     = blockIdx.x / bpb;
    const int chunk = blockIdx.x % bpb;
    if (tid < Pp)     med[tid] = median[b * Pp + tid];
    if (tid < Pp * 3) ls[tid] = 0.0f;
    if (tid < Pp)     lc[tid] = 0u;
    __syncthreads();
    const size_t base = (size_t)b * Nn + (size_t)chunk * (TILE * TPB_STREAM);
    for (int t = 0; t < TPB_STREAM; ++t) {
        size_t tb = base + (size_t)t * TILE;
        if (tid < 32) {
            tdm_load_1d_f32(sx, xyz + tb * 3, TILE3);
            __builtin_amdgcn_s_wait_tensorcnt(0);
        }
        async_ld_b128(&sd[tid * 4], dist + tb + tid * 4);
        async_ld_b128(&sl[tid * 4], lab + tb + tid * 4);
        wait_async0();
        __syncthreads();
        for (int j = 0; j < 4; ++j) {
            int p = tid * 4 + j;
            float d = sd[p];
            int l = sl[p];
            if (d <= med[l]) {
                atomicAdd(&ls[l * 3 + 0], sx[p * 3 + 0]);
                atomicAdd(&ls[l * 3 + 1], sx[p * 3 + 1]);
                atomicAdd(&ls[l * 3 + 2], sx[p * 3 + 2]);
                atomicAdd(&lc[l], 1u);
            }
        }
        __syncthreads();
    }
    if (tid < Pp * 3) atomicAdd(&fsum[b * Pp * 3 + tid], ls[tid]);
    if (tid < Pp)     atomicAdd(&fcnt[b * Pp + tid], lc[tid]);
}

// ---------------------------------------------------------------------------
// Kernel 7: poses = fsum / max(fcnt,1)
// ---------------------------------------------------------------------------
__global__ void k_out(const float* __restrict__ fsum, const unsigned* __restrict__ fcnt,
                      float* __restrict__ out) {
    int g = blockIdx.x * blockDim.x + threadIdx.x;
    if (g >= Gg) return;
    unsigned c = fcnt[g];
    float f = (float)(c ? c : 1u);
    out[g * 3 + 0] = fsum[g * 3 + 0] / f;
    out[g * 3 + 1] = fsum[g * 3 + 1] / f;
    out[g * 3 + 2] = fsum[g * 3 + 2] / f;
}

// ---------------------------------------------------------------------------
// Host orchestration (graph-capture safe: kernel launches + memsetAsync only)
// ---------------------------------------------------------------------------
extern "C" void kernel_launch(void* const* d_in, const int* in_sizes, int n_in,
                              void* d_out, int out_size, void* d_ws, size_t ws_size,
                              hipStream_t stream) {
    const float* xyz = (const float*)d_in[0];
    const int*   lab = (const int*)d_in[1];
    float*       out = (float*)d_out;

    // Workspace carve-up (~35 MB: 1.2 MB hist + 33.5 MB dist cache + small)
    float*    gsum   = (float*)d_ws;                   // G*3
    float*    fsum   = gsum + Gg * 3;                  // G*3
    unsigned* gcnt   = (unsigned*)(fsum + Gg * 3);     // G
    unsigned* fcnt   = gcnt + Gg;                      // G
    float*    center = (float*)(fcnt + Gg);            // G*3
    unsigned* selP   = (unsigned*)(center + Gg * 3);   // G
    unsigned* selK   = selP + Gg;                      // G
    float*    median = (float*)(selK + Gg);            // G
    unsigned* hist   = (unsigned*)(median + Gg);       // G*256
    float*    dist   = (float*)(hist + (size_t)Gg * 256); // B*N

    const int gridStream = Bb * (Nn / (TILE * TPB_STREAM));   // 1024
    const int gridG      = (Gg + 127) / 128;

    // zero accumulators (gsum,fsum,gcnt,fcnt contiguous)
    hipMemsetAsync(gsum, 0, (size_t)Gg * 8 * sizeof(float), stream);

    // 1) segment sums -> centers
    k_segsum<<<gridStream, 256, 0, stream>>>(xyz, lab, gsum, gcnt);
    k_centers<<<gridG, 128, 0, stream>>>(gsum, gcnt, center, selP, selK);

    // 2) distances + 4-level exact radix select of the per-group median
    hipMemsetAsync(hist, 0, (size_t)Gg * 256 * sizeof(unsigned), stream);
    k_dist_hist0<<<gridStream, 256, 0, stream>>>(xyz, lab, center, dist, hist);
    k_scan<<<gridG, 128, 0, stream>>>(hist, selP, selK, median, 23, 256, 0);

    hipMemsetAsync(hist, 0, (size_t)Gg * 256 * sizeof(unsigned), stream);
    k_hist_refine<<<gridStream, 256, 0, stream>>>(dist, lab, selP, hist, 15, 23, 0xFFu);
    k_scan<<<gridG, 128, 0, stream>>>(hist, selP, selK, median, 15, 256, 0);

    hipMemsetAsync(hist, 0, (size_t)Gg * 256 * sizeof(unsigned), stream);
    k_hist_refine<<<gridStream, 256, 0, stream>>>(dist, lab, selP, hist, 7, 15, 0xFFu);
    k_scan<<<gridG, 128, 0, stream>>>(hist, selP, selK, median, 7, 256, 0);

    hipMemsetAsync(hist, 0, (size_t)Gg * 256 * sizeof(unsigned), stream);
    k_hist_refine<<<gridStream, 256, 0, stream>>>(dist, lab, selP, hist, 0, 7, 0x7Fu);
    k_scan<<<gridG, 128, 0, stream>>>(hist, selP, selK, median, 0, 128, 1);

    // 3) filtered means -> poses
    k_final<<<gridStream, 256, 0, stream>>>(xyz, dist, lab, median, fsum, fcnt);
    k_out<<<gridG, 128, 0, stream>>>(fsum, fcnt, out);
}